// Encoder_65661460021577
// MI455X (gfx1250) — compile-verified
//
#include <hip/hip_runtime.h>

typedef __attribute__((ext_vector_type(2))) float v2f;
typedef __attribute__((ext_vector_type(4))) float v4f;
typedef __attribute__((ext_vector_type(8))) float v8f;

#define B_   4
#define N_   4096
#define C_   256
#define NT_  (N_ / 16)   // 256 sixteen-wide tiles per dimension

#define WMMA_F32(Aop, Bop, Cop)                                              \
  __builtin_amdgcn_wmma_f32_16x16x4_f32(false, (Aop), false, (Bop),          \
                                        (short)0, (Cop), false, false)

// Sorted-ascending 8-entry register heap insert (common path: 1 compare).
__device__ __forceinline__ void insert8(float (&h)[8], float v) {
  if (v < h[7]) {
    float x = v;
    #pragma unroll
    for (int s = 0; s < 8; ++s) {
      float mn = fminf(h[s], x);
      float mx = fmaxf(h[s], x);
      h[s] = mn; x = mx;
    }
  }
}

// ---------------------------------------------------------------------------
// Kernel 1a: per-point squared norms. One wave per point, 8 floats per lane.
// ---------------------------------------------------------------------------
__global__ void sqnorm_kernel(const float* __restrict__ feat,
                              float* __restrict__ sq) {
  int wave = threadIdx.x >> 5;
  int lane = threadIdx.x & 31;
  int p = blockIdx.x * 8 + wave;
  const float4* f4 = (const float4*)(feat + (size_t)p * C_);
  float4 a = f4[lane * 2 + 0];
  float4 b = f4[lane * 2 + 1];
  float s = a.x * a.x + a.y * a.y + a.z * a.z + a.w * a.w +
            b.x * b.x + b.y * b.y + b.z * b.z + b.w * b.w;
  #pragma unroll
  for (int m = 16; m >= 1; m >>= 1) s += __shfl_xor(s, m, 32);
  if (lane == 0) sq[p] = s;
}

// ---------------------------------------------------------------------------
// Kernel 1b: squared norms + swizzled copy. swz[p][h][m] holds original
// element k = (m>>1)*4 + 2h + (m&1), so a lane's WMMA operand stream
// (k0+2h, k0+2h+1 for k0 = 0,4,8,...) becomes fully contiguous -> b128 loads.
// ---------------------------------------------------------------------------
__global__ void sqnorm_swz_kernel(const float* __restrict__ feat,
                                  float* __restrict__ sq,
                                  float* __restrict__ swz) {
  int wave = threadIdx.x >> 5;
  int lane = threadIdx.x & 31;
  int p = blockIdx.x * 8 + wave;
  const float4* f4 = (const float4*)(feat + (size_t)p * C_);
  float4 a = f4[lane * 2 + 0];            // k = lane*8 + {0,1,2,3}
  float4 b = f4[lane * 2 + 1];            // k = lane*8 + {4,5,6,7}
  float* row = swz + (size_t)p * C_;
  *(float4*)(row + lane * 4)       = make_float4(a.x, a.y, b.x, b.y);  // h=0
  *(float4*)(row + 128 + lane * 4) = make_float4(a.z, a.w, b.z, b.w);  // h=1
  float s = a.x * a.x + a.y * a.y + a.z * a.z + a.w * a.w +
            b.x * b.x + b.y * b.y + b.z * b.z + b.w * b.w;
  #pragma unroll
  for (int m = 16; m >= 1; m >>= 1) s += __shfl_xor(s, m, 32);
  if (lane == 0) sq[p] = s;
}

// ---------------------------------------------------------------------------
// Kernel 2: streamed distance tiles + per-row 8-NN density.
// Grid: B * (N/16) workgroups of 256 threads (8 waves); wave w sweeps column
// tiles w, w+8, ... A (row block) is register-cached across all 256 tiles.
// B loads are issued in groups with distinct registers so one s_wait covers
// 8 WMMAs. SWZ=true: b128 operand loads from the swizzled copy.
// ---------------------------------------------------------------------------
template <bool SWZ>
__global__ void __launch_bounds__(256)
dps_density_kernel(const float* __restrict__ data,   // swz copy or raw feat
                   const float* __restrict__ sq,
                   float* __restrict__ density) {
  __shared__ float cand[8][8][16][8];   // [row][wave][lane lo][slot] = 32 KB

  int b = blockIdx.x / NT_;
  int rowbase = (blockIdx.x % NT_) * 16;
  int wave = threadIdx.x >> 5;
  int lane = threadIdx.x & 31;
  int lo = lane & 15;                    // tile row/col owned by this lane
  int hi = lane >> 4;                    // K-half selector

  const float* SQ = sq + b * N_;

  float h[8][8];
  #pragma unroll
  for (int r = 0; r < 8; ++r)
    #pragma unroll
    for (int s = 0; s < 8; ++s) h[r][s] = 3.4e38f;

  float sq_r[8];
  #pragma unroll
  for (int r = 0; r < 8; ++r) sq_r[r] = SQ[rowbase + r + 8 * hi];

  // Register-cache the A operands for the whole row block (128 VGPRs).
  v4f a4[32];
  v2f a2[64];
  if constexpr (SWZ) {
    const v4f* A4 =
        (const v4f*)(data + ((size_t)(b * N_ + rowbase + lo) * 2 + hi) * 128);
    #pragma unroll
    for (int i = 0; i < 32; ++i) a4[i] = A4[i];
  } else {
    const float* Arow = data + (size_t)(b * N_ + rowbase + lo) * C_ + 2 * hi;
    #pragma unroll
    for (int i = 0; i < 64; ++i) a2[i] = *(const v2f*)(Arow + i * 4);
  }

  for (int jt = wave; jt < NT_; jt += 8) {
    int colbase = jt * 16;
    v8f c = {0.f, 0.f, 0.f, 0.f, 0.f, 0.f, 0.f, 0.f};

    if constexpr (SWZ) {
      const v4f* B4 =
          (const v4f*)(data + ((size_t)(b * N_ + colbase + lo) * 2 + hi) * 128);
      #pragma unroll
      for (int g = 0; g < 32; g += 4) {   // 4 b128 loads -> 8 WMMAs
        v4f b0 = B4[g + 0];
        v4f b1 = B4[g + 1];
        v4f b2 = B4[g + 2];
        v4f b3 = B4[g + 3];
        c = WMMA_F32(a4[g + 0].xy, b0.xy, c);
        c = WMMA_F32(a4[g + 0].zw, b0.zw, c);
        c = WMMA_F32(a4[g + 1].xy, b1.xy, c);
        c = WMMA_F32(a4[g + 1].zw, b1.zw, c);
        c = WMMA_F32(a4[g + 2].xy, b2.xy, c);
        c = WMMA_F32(a4[g + 2].zw, b2.zw, c);
        c = WMMA_F32(a4[g + 3].xy, b3.xy, c);
        c = WMMA_F32(a4[g + 3].zw, b3.zw, c);
      }
    } else {
      const float* Brow =
          data + (size_t)(b * N_ + colbase + lo) * C_ + 2 * hi;
      #pragma unroll
      for (int g = 0; g < 64; g += 8) {   // 8 b64 loads -> 8 WMMAs
        v2f b0 = *(const v2f*)(Brow + (g + 0) * 4);
        v2f b1 = *(const v2f*)(Brow + (g + 1) * 4);
        v2f b2 = *(const v2f*)(Brow + (g + 2) * 4);
        v2f b3 = *(const v2f*)(Brow + (g + 3) * 4);
        v2f b4 = *(const v2f*)(Brow + (g + 4) * 4);
        v2f b5 = *(const v2f*)(Brow + (g + 5) * 4);
        v2f b6 = *(const v2f*)(Brow + (g + 6) * 4);
        v2f b7 = *(const v2f*)(Brow + (g + 7) * 4);
        c = WMMA_F32(a2[g + 0], b0, c);
        c = WMMA_F32(a2[g + 1], b1, c);
        c = WMMA_F32(a2[g + 2], b2, c);
        c = WMMA_F32(a2[g + 3], b3, c);
        c = WMMA_F32(a2[g + 4], b4, c);
        c = WMMA_F32(a2[g + 5], b5, c);
        c = WMMA_F32(a2[g + 6], b6, c);
        c = WMMA_F32(a2[g + 7], b7, c);
      }
    }

    float sqc = SQ[colbase + lo];
    #pragma unroll
    for (int r = 0; r < 8; ++r)
      insert8(h[r], fmaxf(sq_r[r] + sqc - 2.0f * c[r], 0.0f));
  }

  // Merge 8 waves x 16 lanes per row; two passes (rows 0-7, then 8-15).
  for (int pass = 0; pass < 2; ++pass) {
    __syncthreads();
    if (hi == pass) {
      #pragma unroll
      for (int r = 0; r < 8; ++r)
        #pragma unroll
        for (int s = 0; s < 8; ++s) cand[r][wave][lo][s] = h[r][s];
    }
    __syncthreads();
    if (threadIdx.x < 8) {
      int r = threadIdx.x;
      float best[8];
      #pragma unroll
      for (int s = 0; s < 8; ++s) best[s] = 3.4e38f;
      const float* row = &cand[r][0][0][0];
      for (int g = 0; g < 128; ++g) {          // 128 sorted 8-lists
        const float* lst = row + g * 8;
        #pragma unroll
        for (int s = 0; s < 8; ++s) {
          float v = lst[s];
          if (v >= best[7]) break;             // ascending lists -> early out
          float x = v;
          #pragma unroll
          for (int t = 0; t < 8; ++t) {
            float mn = fminf(best[t], x);
            float mx = fmaxf(best[t], x);
            best[t] = mn; x = mx;
          }
        }
      }
      float acc = 0.f;
      #pragma unroll
      for (int s = 0; s < 8; ++s) acc += sqrtf(best[s]);
      density[b * N_ + rowbase + pass * 8 + r] = acc * 0.125f;
    }
  }
}

// ---------------------------------------------------------------------------
// Kernel 3: stable top-M by rank counting (matches jax.lax.top_k ordering).
// ---------------------------------------------------------------------------
__global__ void rank_kernel(const float* __restrict__ density,
                            int* __restrict__ inds, int M) {
  int b = blockIdx.x >> 4;
  int i = ((blockIdx.x & 15) << 8) + threadIdx.x;
  const float* d = density + b * N_;
  float di = d[i];
  int rank = 0;
  for (int j = 0; j < N_; ++j) {
    float dj = d[j];
    rank += (dj > di) || (dj == di && j < i);
  }
  if (rank < M) inds[b * M + rank] = i;
}

// ---------------------------------------------------------------------------
// Kernel 4: gather sampled features (float4) and pos.
// ---------------------------------------------------------------------------
__global__ void gather_kernel(const float* __restrict__ feat,
                              const float* __restrict__ pos,
                              const int* __restrict__ inds,
                              float* __restrict__ outF,
                              float* __restrict__ outP, int M) {
  int b = blockIdx.x / M;
  int m = blockIdx.x % M;
  int i = inds[b * M + m];
  const float4* src = (const float4*)(feat + ((size_t)b * N_ + i) * C_);
  float4* dst = (float4*)(outF + ((size_t)b * M + m) * C_);
  dst[threadIdx.x] = src[threadIdx.x];
  if (threadIdx.x < 3)
    outP[((size_t)b * M + m) * 3 + threadIdx.x] =
        pos[((size_t)b * N_ + i) * 3 + threadIdx.x];
}

// ---------------------------------------------------------------------------
extern "C" void kernel_launch(void* const* d_in, const int* in_sizes, int n_in,
                              void* d_out, int out_size, void* d_ws,
                              size_t ws_size, hipStream_t stream) {
  const float* feat = (const float*)d_in[0];
  const float* pos  = (const float*)d_in[1];
  int M = out_size / (B_ * (C_ + 3));            // 819 for subsample_factor=5

  size_t swzFloats   = (size_t)B_ * N_ * C_;     // 16.8 MB swizzled copy
  size_t smallBytes  = (size_t)(2 * B_ * N_) * sizeof(float) +
                       (size_t)B_ * M * sizeof(int);
  bool use_swz = ws_size >= swzFloats * sizeof(float) + smallBytes;

  float* base    = (float*)d_ws;
  float* swz     = base;
  float* sq      = use_swz ? base + swzFloats : base;
  float* density = sq + B_ * N_;
  int*   inds    = (int*)(density + B_ * N_);
  float* outF = (float*)d_out;
  float* outP = outF + (size_t)B_ * M * C_;

  if (use_swz) {
    sqnorm_swz_kernel<<<(B_ * N_) / 8, 256, 0, stream>>>(feat, sq, swz);
    dps_density_kernel<true><<<B_ * NT_, 256, 0, stream>>>(swz, sq, density);
  } else {
    sqnorm_kernel<<<(B_ * N_) / 8, 256, 0, stream>>>(feat, sq);
    dps_density_kernel<false><<<B_ * NT_, 256, 0, stream>>>(feat, sq, density);
  }
  rank_kernel<<<B_ * 16, 256, 0, stream>>>(density, inds, M);
  gather_kernel<<<B_ * M, 64, 0, stream>>>(feat, pos, inds, outF, outP, M);
}